// DSTBranch_19138374271253
// MI455X (gfx1250) — compile-verified
//
#include <hip/hip_runtime.h>
#include <cmath>

typedef float v2f __attribute__((ext_vector_type(2)));
typedef float v8f __attribute__((ext_vector_type(8)));

#define NPTS 32768      // 32^3 tokens
#define CF   64
#define MAXM 512        // actual gathered counts are ~280/64/139

// D(16x16,f32) += A(16x4,f32) x B(4x16,f32), full fp32 WMMA (CDNA5)
__device__ __forceinline__ v8f wmma4(v2f a, v2f b, v8f c) {
  return __builtin_amdgcn_wmma_f32_16x16x4_f32(false, a, false, b, (short)0, c,
                                               false, false);
}

// ---------------- patch embed: t[n][c] = conv(x) + bias + pos ----------------
__global__ void patch_embed_kernel(const float* __restrict__ x,
                                   const float* __restrict__ w,   // [64][64]
                                   const float* __restrict__ bias,
                                   const float* __restrict__ pos, // [64][32768]
                                   float* __restrict__ t) {       // [32768][64]
  const int ntile = blockIdx.x;          // 0..2047
  const int ctile = blockIdx.y;          // 0..3
  const int lane  = threadIdx.x;
  const int hi    = lane >> 4;           // K half select
  const int l16   = lane & 15;

  const int n  = ntile * 16 + l16;       // token (A row)
  const int vz = n >> 10, vy = (n >> 5) & 31, vx = n & 31;
  const float* px = x + (((vz * 4) * 128 + (vy * 4)) * 128 + vx * 4);
  const int col = ctile * 16 + l16;      // channel (B col)
  const float* brow = w + col * 64;

  v8f acc = {};
#pragma unroll
  for (int kb = 0; kb < 64; kb += 4) {
    const int k0 = kb + hi * 2;
    v2f a, b;
    { int kz = k0 >> 4, ky = (k0 >> 2) & 3, kx = k0 & 3;
      a.x = px[(kz * 128 + ky) * 128 + kx]; }
    { int k1 = k0 + 1; int kz = k1 >> 4, ky = (k1 >> 2) & 3, kx = k1 & 3;
      a.y = px[(kz * 128 + ky) * 128 + kx]; }
    b.x = brow[k0];
    b.y = brow[k0 + 1];
    acc = wmma4(a, b, acc);
  }
#pragma unroll
  for (int v = 0; v < 8; v++) {
    const int mm = ntile * 16 + v + hi * 8;   // D row
    t[mm * 64 + col] = acc[v] + bias[col] + pos[col * NPTS + mm];
  }
}

// ---------------- deterministic ordered compaction of dilation mask ---------
__global__ void build_idx_kernel(int dil, int* __restrict__ idx, int Mcap) {
  __shared__ int base;
  __shared__ int wsum[32];
  __shared__ int woff[32];
  if (threadIdx.x == 0) base = 0;
  __syncthreads();
  for (int chunk = 0; chunk < NPTS; chunk += 1024) {
    const int p  = chunk + (int)threadIdx.x;
    const int z  = p >> 10, y = (p >> 5) & 31, x = p & 31;
    const int dz = z - 16, dy = y - 16, dx = x - 16;
    const int s  = dz * dz + dy * dy + dx * dx;
    const int k  = (int)(sqrtf((float)s) + 0.5f);
    const int flag = (k * k == s && (k % dil) == 0) ? 1 : 0;

    const unsigned mask = (unsigned)__ballot(flag);
    const int lane = threadIdx.x & 31, wid = threadIdx.x >> 5;
    const int prefix = __popc(mask & ((1u << lane) - 1u));
    if (lane == 0) wsum[wid] = __popc(mask);
    __syncthreads();
    if (threadIdx.x == 0) {
      int acc = base;
      for (int i = 0; i < 32; i++) { woff[i] = acc; acc += wsum[i]; }
      base = acc;
    }
    __syncthreads();
    if (flag) {
      const int o = woff[wid] + prefix;
      if (o < Mcap) idx[o] = p;
    }
    __syncthreads();
  }
}

// ---------------- qkv at gathered rows: qs[m][j] = t[idx[m]] @ qkv_w^T + b ---
__global__ void qkv_gather_kernel(const float* __restrict__ t,
                                  const int* __restrict__ idx,
                                  const float* __restrict__ qw,  // [192][64]
                                  const float* __restrict__ qb,
                                  float* __restrict__ qs,        // [Mpad][192]
                                  int M) {
  const int mtile = blockIdx.x, jtile = blockIdx.y;  // jtile 0..11
  const int lane = threadIdx.x, hi = lane >> 4, l16 = lane & 15;
  const int m = mtile * 16 + l16;
  const int row = idx[min(m, M - 1)];                // clamp pad rows (finite)
  const float* arow = t + row * 64;
  const int col = jtile * 16 + l16;
  const float* brow = qw + col * 64;

  v8f acc = {};
#pragma unroll
  for (int kb = 0; kb < 64; kb += 4) {
    const int k0 = kb + hi * 2;
    v2f a, b;
    a.x = arow[k0];  a.y = arow[k0 + 1];
    b.x = brow[k0];  b.y = brow[k0 + 1];
    acc = wmma4(a, b, acc);
  }
#pragma unroll
  for (int v = 0; v < 8; v++) {
    const int mm = mtile * 16 + v + hi * 8;
    qs[mm * 192 + col] = acc[v] + qb[col];
  }
}

// ---------------- S[m][n] = scale * (q_m . k_n); zero outside MxM -----------
__global__ void scores_kernel(const float* __restrict__ qs,
                              float* __restrict__ S, int M, int Sp, float scale) {
  const int mtile = blockIdx.x, ntile = blockIdx.y;
  const int lane = threadIdx.x, hi = lane >> 4, l16 = lane & 15;
  const float* arow = qs + (mtile * 16 + l16) * 192;        // q
  const float* brow = qs + (ntile * 16 + l16) * 192 + 64;   // k (B col)

  v8f acc = {};
#pragma unroll
  for (int kb = 0; kb < 64; kb += 4) {
    const int k0 = kb + hi * 2;
    v2f a, b;
    a.x = arow[k0];  a.y = arow[k0 + 1];
    b.x = brow[k0];  b.y = brow[k0 + 1];
    acc = wmma4(a, b, acc);
  }
  const int nn = ntile * 16 + l16;
#pragma unroll
  for (int v = 0; v < 8; v++) {
    const int mm = mtile * 16 + v + hi * 8;
    S[mm * Sp + nn] = (mm < M && nn < M) ? acc[v] * scale : 0.0f;
  }
}

// ---------------- row softmax over first M columns (in place) ---------------
__global__ void softmax_rows_kernel(float* __restrict__ S, int M, int Sp) {
  const int m = blockIdx.x;
  float* row = S + (size_t)m * Sp;
  __shared__ float red[256];
  float mx = -INFINITY;
  for (int n = threadIdx.x; n < M; n += 256) mx = fmaxf(mx, row[n]);
  red[threadIdx.x] = mx;  __syncthreads();
  for (int s = 128; s > 0; s >>= 1) {
    if (threadIdx.x < s) red[threadIdx.x] = fmaxf(red[threadIdx.x], red[threadIdx.x + s]);
    __syncthreads();
  }
  mx = red[0];  __syncthreads();
  float sum = 0.0f;
  for (int n = threadIdx.x; n < M; n += 256) {
    const float e = expf(row[n] - mx);
    row[n] = e;  sum += e;
  }
  red[threadIdx.x] = sum;  __syncthreads();
  for (int s = 128; s > 0; s >>= 1) {
    if (threadIdx.x < s) red[threadIdx.x] += red[threadIdx.x + s];
    __syncthreads();
  }
  const float inv = 1.0f / red[0];
  for (int n = threadIdx.x; n < M; n += 256) row[n] *= inv;
}

// ---------------- y[m][c] = P[m][:] @ V[:,c]  (K = Mpad, pads are 0) --------
__global__ void pv_kernel(const float* __restrict__ S,
                          const float* __restrict__ qs,
                          float* __restrict__ yb, int Kp, int Sp) {
  const int mtile = blockIdx.x, ctile = blockIdx.y;   // ctile 0..3
  const int lane = threadIdx.x, hi = lane >> 4, l16 = lane & 15;
  const float* arow = S + (size_t)(mtile * 16 + l16) * Sp;
  const int col = ctile * 16 + l16;

  v8f acc = {};
  for (int kb = 0; kb < Kp; kb += 4) {
    const int k0 = kb + hi * 2;
    v2f a, b;
    a.x = arow[k0];                    a.y = arow[k0 + 1];
    b.x = qs[k0 * 192 + 128 + col];    b.y = qs[(k0 + 1) * 192 + 128 + col];
    acc = wmma4(a, b, acc);
  }
#pragma unroll
  for (int v = 0; v < 8; v++) {
    const int mm = mtile * 16 + v + hi * 8;
    yb[mm * 64 + col] = acc[v];
  }
}

// ---------------- out = y @ proj_w^T + proj_b + y; scatter to t[idx] --------
__global__ void proj_scatter_kernel(const float* __restrict__ yb,
                                    const float* __restrict__ pw, // [64][64]
                                    const float* __restrict__ pb,
                                    const int* __restrict__ idx,
                                    float* __restrict__ t, int M) {
  const int mtile = blockIdx.x, ctile = blockIdx.y;
  const int lane = threadIdx.x, hi = lane >> 4, l16 = lane & 15;
  const float* arow = yb + (mtile * 16 + l16) * 64;
  const int col = ctile * 16 + l16;
  const float* brow = pw + col * 64;

  v8f acc = {};
#pragma unroll
  for (int kb = 0; kb < 64; kb += 4) {
    const int k0 = kb + hi * 2;
    v2f a, b;
    a.x = arow[k0];  a.y = arow[k0 + 1];
    b.x = brow[k0];  b.y = brow[k0 + 1];
    acc = wmma4(a, b, acc);
  }
#pragma unroll
  for (int v = 0; v < 8; v++) {
    const int mm = mtile * 16 + v + hi * 8;
    if (mm < M) {
      const int row = idx[mm];
      t[row * 64 + col] = acc[v] + pb[col] + yb[mm * 64 + col];
    }
  }
}

// ---------------- LDS-tiled transpose: t[n][c] -> h[c][n] -------------------
// Makes the upsample's gathers unit-stride in x (h is channel-major).
__global__ void transpose_kernel(const float* __restrict__ t,
                                 float* __restrict__ h) {
  __shared__ float tile[16][17];                 // +1 pad: bank-conflict free
  const int nb = blockIdx.x * 16, cb = blockIdx.y * 16;
  const int tx = threadIdx.x, ty = threadIdx.y;
  tile[ty][tx] = t[(nb + ty) * 64 + cb + tx];    // coalesced read (c innermost)
  __syncthreads();
  h[(cb + ty) * NPTS + nb + tx] = tile[tx][ty];  // coalesced write (n innermost)
}

// ---------------- trilinear x4 upsample: coalesced loads + HBM stores -------
__global__ void upsample_kernel(const float* __restrict__ h,  // [64][32][32][32]
                                float* __restrict__ out) {
  const unsigned gid = blockIdx.x * 256u + threadIdx.x;
  const int xo = gid & 127, yo = (gid >> 7) & 127, zo = (gid >> 14) & 127;
  const int c = gid >> 21;
  const float r = (float)(31.0 / 127.0);

  float pz = zo * r; int z0 = (int)pz; if (z0 > 31) z0 = 31;
  float wz = pz - (float)z0; int z1 = z0 + 1; if (z1 > 31) z1 = 31;
  float py = yo * r; int y0 = (int)py; if (y0 > 31) y0 = 31;
  float wy = py - (float)y0; int y1 = y0 + 1; if (y1 > 31) y1 = 31;
  float pxp = xo * r; int x0 = (int)pxp; if (x0 > 31) x0 = 31;
  float wx = pxp - (float)x0; int x1 = x0 + 1; if (x1 > 31) x1 = 31;

  const float* hc = h + (size_t)c * NPTS;
  auto ld = [&](int zi, int yi, int xi) {
    return hc[(zi * 32 + yi) * 32 + xi];         // unit stride in xi
  };
  const float a00 = ld(z0, y0, x0) * (1.f - wz) + ld(z1, y0, x0) * wz;
  const float a10 = ld(z0, y1, x0) * (1.f - wz) + ld(z1, y1, x0) * wz;
  const float a01 = ld(z0, y0, x1) * (1.f - wz) + ld(z1, y0, x1) * wz;
  const float a11 = ld(z0, y1, x1) * (1.f - wz) + ld(z1, y1, x1) * wz;
  const float b0 = a00 * (1.f - wy) + a10 * wy;
  const float b1 = a01 * (1.f - wy) + a11 * wy;
  out[gid] = b0 * (1.f - wx) + b1 * wx;
}

// ---------------- host: identical integer mask logic (pure, deterministic) --
static inline bool mask_point(int p, int dil) {
  const int z = p >> 10, y = (p >> 5) & 31, x = p & 31;
  const int dz = z - 16, dy = y - 16, dx = x - 16;
  const int s = dz * dz + dy * dy + dx * dx;
  const int k = (int)(sqrtf((float)s) + 0.5f);
  return (k * k == s) && (k % dil == 0);
}

extern "C" void kernel_launch(void* const* d_in, const int* in_sizes, int n_in,
                              void* d_out, int out_size, void* d_ws, size_t ws_size,
                              hipStream_t stream) {
  (void)in_sizes; (void)n_in; (void)out_size; (void)ws_size;
  const float* x   = (const float*)d_in[0];
  const float* wp  = (const float*)d_in[1];
  const float* bp  = (const float*)d_in[2];
  const float* pos = (const float*)d_in[3];
  const float* qkvw[3] = {(const float*)d_in[4], (const float*)d_in[8],  (const float*)d_in[12]};
  const float* qkvb[3] = {(const float*)d_in[5], (const float*)d_in[9],  (const float*)d_in[13]};
  const float* prjw[3] = {(const float*)d_in[6], (const float*)d_in[10], (const float*)d_in[14]};
  const float* prjb[3] = {(const float*)d_in[7], (const float*)d_in[11], (const float*)d_in[15]};
  float* out = (float*)d_out;

  float* t    = (float*)d_ws;                  // 32768*64  (8 MB)
  float* h    = t  + (size_t)NPTS * CF;        // 64*32768  (8 MB, channel-major)
  float* qs   = h  + (size_t)NPTS * CF;        // MAXM*192
  float* S    = qs + (size_t)MAXM * 192;       // MAXM*MAXM (1 MB)
  float* yb   = S  + (size_t)MAXM * MAXM;      // MAXM*64
  int*   idxb = (int*)(yb + (size_t)MAXM * 64);

  patch_embed_kernel<<<dim3(NPTS / 16, 4), 32, 0, stream>>>(x, wp, bp, pos, t);

  const int dils[3] = {2, 4, 6};
  for (int b = 0; b < 3; b++) {
    const int dil = dils[b];
    int M = 0;
    for (int p = 0; p < NPTS; p++) M += mask_point(p, dil) ? 1 : 0;
    if (M > MAXM) M = MAXM;
    const int Mpad = (M + 15) & ~15;
    const int tm = Mpad / 16;

    build_idx_kernel<<<1, 1024, 0, stream>>>(dil, idxb, MAXM);
    qkv_gather_kernel<<<dim3(tm, 12), 32, 0, stream>>>(t, idxb, qkvw[b], qkvb[b], qs, M);
    scores_kernel<<<dim3(tm, tm), 32, 0, stream>>>(qs, S, M, Mpad, 0.125f);
    softmax_rows_kernel<<<M, 256, 0, stream>>>(S, M, Mpad);
    pv_kernel<<<dim3(tm, 4), 32, 0, stream>>>(S, qs, yb, Mpad, Mpad);
    proj_scatter_kernel<<<dim3(tm, 4), 32, 0, stream>>>(yb, prjw[b], prjb[b], idxb, t, M);
  }

  transpose_kernel<<<dim3(NPTS / 16, CF / 16), dim3(16, 16), 0, stream>>>(t, h);
  upsample_kernel<<<(64u * 128u * 128u * 128u) / 256u, 256, 0, stream>>>(h, out);
}